// MeanAggregator_10368051053026
// MI455X (gfx1250) — compile-verified
//
#include <hip/hip_runtime.h>

#define NUM_SAMPLE 10
#define D_FEAT     128
#define WAVES_PER_BLOCK 8
#define THREADS (WAVES_PER_BLOCK * 32)
#define ROW_BYTES (D_FEAT * 4)          // 512 B per feature row
#define BUF_FLOATS (NUM_SAMPLE * D_FEAT)

// Batch-load the 10 row indices (wide s_load, single kmcnt wait), then fire
// all 10 async b128 gathers into this wave's LDS buffer back-to-back.
__device__ __forceinline__ void issue_row_gather(const long long* __restrict__ ip,
                                                 unsigned long long featBase,
                                                 unsigned laneByte,
                                                 unsigned ldsBaseLane) {
    long long idxs[NUM_SAMPLE];
#pragma unroll
    for (int s = 0; s < NUM_SAMPLE; ++s) idxs[s] = ip[s];   // uniform -> SGPRs
#pragma unroll
    for (int s = 0; s < NUM_SAMPLE; ++s) {
        const unsigned long long gaddr =
            featBase + (((unsigned long long)idxs[s]) << 9) + laneByte;  // idx*512B
        const unsigned laddr = ldsBaseLane + (unsigned)(s * ROW_BYTES);
        asm volatile("global_load_async_to_lds_b128 %0, %1, off"
                     :: "v"(laddr), "v"(gaddr)
                     : "memory");
    }
}

// One wave32 per output row; each wave walks ~10 rows with a double-buffered
// async-LDS pipeline: row i+1 streams into one buffer (ASYNCcnt) while row i
// is reduced out of the other (DScnt).
__global__ __launch_bounds__(THREADS)
void MeanAggregator_10368051053026_kernel(const float* __restrict__ feat,
                                          const long long* __restrict__ nidx,
                                          float* __restrict__ out,
                                          int nrows) {
    // 8 waves * 2 buffers * 10 rows * 512B = 80 KB per block (4 blocks / WGP)
    __shared__ __align__(16) float smem[WAVES_PER_BLOCK * 2 * BUF_FLOATS];

    const int lane = threadIdx.x & 31;
    const int wave = threadIdx.x >> 5;
    const int wgFlat = __builtin_amdgcn_readfirstlane((int)blockIdx.x * WAVES_PER_BLOCK + wave);
    const int stride = (int)gridDim.x * WAVES_PER_BLOCK;

    int row = wgFlat;
    if (row >= nrows) return;

    const unsigned laneByte = (unsigned)(lane << 4);
    float* buf0 = &smem[wave * (2 * BUF_FLOATS)];
    float* buf1 = buf0 + BUF_FLOATS;
    // Low 32 bits of the generic shared pointer == LDS byte offset on gfx1250.
    const unsigned off0 = (unsigned)(size_t)(void*)buf0 + laneByte;
    const unsigned off1 = (unsigned)(size_t)(void*)buf1 + laneByte;
    const unsigned long long featBase = (unsigned long long)(const void*)feat;

    // Prologue: start row 0's gathers into buf0.
    issue_row_gather(nidx + (size_t)row * NUM_SAMPLE, featBase, laneByte, off0);

    int parity = 0;  // which buffer holds the current row
    for (;;) {
        const int next = row + stride;
        const bool hasNext = next < nrows;

        if (hasNext) {
            // WAR guard: the buffer we are about to overwrite was read via
            // ds_loads last iteration; async LDS writes are unordered with
            // DS-indexed reads, so drain DScnt before recycling it.
            asm volatile("s_wait_dscnt 0x0" ::: "memory");
            issue_row_gather(nidx + (size_t)next * NUM_SAMPLE, featBase, laneByte,
                             parity ? off0 : off1);
            // Async-load completion is in-order: <=10 outstanding means the
            // current row's 10 loads have landed in LDS.
            asm volatile("s_wait_asynccnt 0xA" ::: "memory");
        } else {
            asm volatile("s_wait_asynccnt 0x0" ::: "memory");
        }

        // Reduce current row: 10 x ds_load_b128 per lane, sum, scale, store.
        const float4* __restrict__ sp = (const float4*)(parity ? buf1 : buf0);
        float4 acc = sp[lane];
#pragma unroll
        for (int s = 1; s < NUM_SAMPLE; ++s) {
            const float4 v = sp[s * 32 + lane];
            acc.x += v.x; acc.y += v.y; acc.z += v.z; acc.w += v.w;
        }
        const float inv = 1.0f / (float)NUM_SAMPLE;
        float4 r;
        r.x = acc.x * inv; r.y = acc.y * inv; r.z = acc.z * inv; r.w = acc.w * inv;

        float4* __restrict__ op = (float4*)(out + (size_t)row * D_FEAT);
        op[lane] = r;

        if (!hasNext) return;
        row = next;
        parity ^= 1;
    }
}

extern "C" void kernel_launch(void* const* d_in, const int* in_sizes, int n_in,
                              void* d_out, int out_size, void* d_ws, size_t ws_size,
                              hipStream_t stream) {
    const float*     feat = (const float*)d_in[0];      // [N_NODES, 128] f32
    const long long* nidx = (const long long*)d_in[1];  // [BATCH, 10] i64
    float*           out  = (float*)d_out;              // [BATCH, 128] f32

    const int nrows = in_sizes[1] / NUM_SAMPLE;         // BATCH
    // ~10 rows per wave: ceil(nrows/10) waves -> /8 waves-per-block blocks.
    int blocks = (nrows + (WAVES_PER_BLOCK * NUM_SAMPLE) - 1) / (WAVES_PER_BLOCK * NUM_SAMPLE);
    if (blocks < 1) blocks = 1;

    MeanAggregator_10368051053026_kernel<<<blocks, THREADS, 0, stream>>>(
        feat, nidx, out, nrows);
}